// LiGhTPredictor_12730283066009
// MI455X (gfx1250) — compile-verified
//
#include <hip/hip_runtime.h>
#include <hip/hip_bf16.h>

typedef float v2f __attribute__((ext_vector_type(2)));
typedef float v8f __attribute__((ext_vector_type(8)));

// ---------------- workspace layout (floats) ----------------
// RR : 320*256  (rows 0..159 = atom_tables@P_top, 160..319 = atom_tables@P_bot)
// Q  :  40*256  (rows 0..19  = charge_W@P_top,    20..39   = charge_W@P_bot)
// BT :   6*256  (dist_table @ bond_proj_W + bond_proj_b)
// AB :     256  (atom_proj_b + charge_b folded through P)
#define WS_RR 0
#define WS_Q  (320*256)
#define WS_BT (360*256)
#define WS_AB (366*256)

// ---------- prep: 6x256 bond lookup table ----------
__global__ void prep_bond_table(const float* __restrict__ dist_table,
                                const float* __restrict__ W,
                                const float* __restrict__ b,
                                float* __restrict__ bt) {
    int i = blockIdx.x, c = threadIdx.x;            // 6 blocks x 256
    float acc = b[c];
    for (int k = 0; k < 128; ++k)
        acc = fmaf(dist_table[i*128 + k], W[k*256 + c], acc);
    bt[i*256 + c] = acc;
}

// ---------- prep: fold vocab tables & charge_W through atom_proj_W ----------
__global__ void prep_atom(const float* __restrict__ at,   // (5,32,128)
                          const float* __restrict__ cW,   // (20,128)
                          const float* __restrict__ cb,   // (128,)
                          const float* __restrict__ P,    // (256,256)
                          const float* __restrict__ pb,   // (256,)
                          float* __restrict__ ws) {
    int ro = blockIdx.x, c = threadIdx.x;           // 361 blocks x 256
    if (ro < 320) {
        int j = ro / 160;                           // 0 = top half of P, 1 = bottom
        const float* src = at + (ro - j*160) * 128;
        int poff = j * 128;
        float acc = 0.f;
        for (int k = 0; k < 128; ++k)
            acc = fmaf(src[k], P[(poff + k)*256 + c], acc);
        ws[WS_RR + ro*256 + c] = acc;
    } else if (ro < 360) {
        int q = ro - 320;
        int j = q / 20;
        const float* src = cW + (q - j*20) * 128;
        int poff = j * 128;
        float acc = 0.f;
        for (int k = 0; k < 128; ++k)
            acc = fmaf(src[k], P[(poff + k)*256 + c], acc);
        ws[WS_Q + q*256 + c] = acc;
    } else {
        float acc = pb[c];                          // fold charge_b through both halves
        for (int k = 0; k < 128; ++k)
            acc = fmaf(cb[k], P[k*256 + c] + P[(128 + k)*256 + c], acc);
        ws[WS_AB + c] = acc;
    }
}

// ---------- atom_h: gather-sum + K=40 charge-RBF rank update ----------
__global__ __launch_bounds__(256) void atom_kernel(const int* __restrict__ atom_cat,
                                                   const float* __restrict__ pcharge,
                                                   const float* __restrict__ ws,
                                                   float* __restrict__ out) {
    __shared__ float s_rbf[4][40];
    __shared__ int   s_off[4][10];
    const float* RR = ws + WS_RR;
    const float* Q  = ws + WS_Q;
    const float* AB = ws + WS_AB;
    int tid  = threadIdx.x;
    int row0 = blockIdx.x * 4;
    if (tid < 160) {                                // 40 RBF weights per row
        int r = tid / 40, q = tid - r*40;
        float p = pcharge[(row0 + r)*2 + (q >= 20)];
        float d = p - 0.1f * (float)(q % 20);
        s_rbf[r][q] = __expf(-10.f * d * d);
    }
    if (tid < 40) {                                 // 10 gather offsets per row
        int r = tid / 10, s = tid - r*10;
        int t = s >> 1, j = s & 1;
        int v = atom_cat[(row0 + r)*10 + s];
        s_off[r][s] = (j*160 + t*32 + v) * 256;
    }
    __syncthreads();
    int r  = tid >> 6;                              // wave-uniform row
    int c4 = (tid & 63) * 4;
    float4 acc = *(const float4*)(AB + c4);
    #pragma unroll
    for (int s = 0; s < 10; ++s) {
        const float4 g = *(const float4*)(RR + s_off[r][s] + c4);
        acc.x += g.x; acc.y += g.y; acc.z += g.z; acc.w += g.w;
    }
    #pragma unroll 8
    for (int q = 0; q < 40; ++q) {
        float w = s_rbf[r][q];
        const float4 g = *(const float4*)(Q + q*256 + c4);
        acc.x = fmaf(w, g.x, acc.x); acc.y = fmaf(w, g.y, acc.y);
        acc.z = fmaf(w, g.z, acc.z); acc.w = fmaf(w, g.w, acc.w);
    }
    *(float4*)(out + (row0 + r)*256 + c4) = acc;
}

// ---------- bond_h: pure table gather ----------
__global__ __launch_bounds__(256) void bond_gather(const float* __restrict__ dist,
                                                   const float* __restrict__ bt,
                                                   float* __restrict__ out) {
    int gid = blockIdx.x * 256 + threadIdx.x;
    int e   = gid >> 6;
    int c4  = (gid & 63) * 4;
    float d = dist[e];
    d = fminf(fmaxf(d, 1.0f), 4.999999f);
    int idx = (int)d;                               // 1..4
    *(float4*)(out + e*256 + c4) = *(const float4*)(bt + idx*256 + c4);
}

// ---------- RBF @ W GEMM via V_WMMA_F32_16X16X4_F32 ----------
// A[m][k] = exp(-10*(x[m]-0.1k)^2), W is (KVALID x 256), centers = 0.1*k.
// Block = 64 threads (2 waves), one 16-row M tile, each wave owns 8 N-tiles.
template<int KSTEPS, int KVALID>
__global__ __launch_bounds__(64) void rbf_wmma(const float* __restrict__ x,
                                               const float* __restrict__ W,
                                               const float* __restrict__ bias,
                                               float* __restrict__ out) {
    const int lane = threadIdx.x & 31;
    const int wv   = threadIdx.x >> 5;
    const int m    = lane & 15;
    const int kh   = lane >> 4;                     // which K-pair half this lane holds
    const int mbase = blockIdx.x * 16;
    const float xv = x[mbase + m];
    v8f acc[8] = {};
    #pragma unroll
    for (int ks = 0; ks < KSTEPS; ++ks) {
        const int k0 = ks*4 + kh*2;
        const float c0 = 0.1f * (float)k0;
        const float d0 = xv - c0;
        const float d1 = xv - (c0 + 0.1f);
        v2f a;
        a.x = (k0     < KVALID) ? __expf(-10.f * d0 * d0) : 0.f;
        a.y = (k0 + 1 < KVALID) ? __expf(-10.f * d1 * d1) : 0.f;
        #pragma unroll
        for (int i = 0; i < 8; ++i) {
            const int col = (wv*8 + i)*16 + m;
            v2f b;
            b.x = (k0     < KVALID) ? W[k0*256 + col]       : 0.f;
            b.y = (k0 + 1 < KVALID) ? W[(k0 + 1)*256 + col] : 0.f;
            acc[i] = __builtin_amdgcn_wmma_f32_16x16x4_f32(
                false, a, false, b, (short)0, acc[i], false, false);
        }
    }
    #pragma unroll
    for (int i = 0; i < 8; ++i) {
        const int col = (wv*8 + i)*16 + m;
        const float bv = bias[col];
        #pragma unroll
        for (int r = 0; r < 8; ++r) {
            const int row = mbase + r + 8*kh;       // C/D layout: VGPR r -> M = r + 8*(lane/16)
            out[row*256 + col] = acc[i][r] + bv;
        }
    }
}

extern "C" void kernel_launch(void* const* d_in, const int* in_sizes, int n_in,
                              void* d_out, int out_size, void* d_ws, size_t ws_size,
                              hipStream_t stream) {
    const int*   atom_cat    = (const int*)  d_in[0];
    const float* pcharge     = (const float*)d_in[1];
    const float* dist_feat   = (const float*)d_in[2];
    const float* angle_feat  = (const float*)d_in[3];
    const float* atom_tables = (const float*)d_in[4];
    const float* charge_W    = (const float*)d_in[5];
    const float* charge_b    = (const float*)d_in[6];
    const float* atom_proj_W = (const float*)d_in[7];
    const float* atom_proj_b = (const float*)d_in[8];
    const float* dist_table  = (const float*)d_in[9];
    const float* bond_proj_W = (const float*)d_in[10];
    const float* bond_proj_b = (const float*)d_in[11];
    const float* bond_rbf_W  = (const float*)d_in[12];
    const float* bond_rbf_b  = (const float*)d_in[13];
    const float* angle_rbf_W = (const float*)d_in[14];
    const float* angle_rbf_b = (const float*)d_in[15];
    (void)in_sizes; (void)n_in; (void)out_size; (void)ws_size;

    float* out = (float*)d_out;
    float* ws  = (float*)d_ws;

    float* out_atom  = out;                         // 200000 x 256
    float* out_bond  = out + (size_t)200000 * 256;  // 400000 x 256
    float* out_bondf = out + (size_t)600000 * 256;  // 400000 x 256
    float* out_angle = out + (size_t)1000000 * 256; // 400000 x 256

    prep_bond_table<<<dim3(6),   dim3(256), 0, stream>>>(dist_table, bond_proj_W, bond_proj_b,
                                                         ws + WS_BT);
    prep_atom<<<dim3(361), dim3(256), 0, stream>>>(atom_tables, charge_W, charge_b,
                                                   atom_proj_W, atom_proj_b, ws);

    atom_kernel<<<dim3(50000),  dim3(256), 0, stream>>>(atom_cat, pcharge, ws, out_atom);
    bond_gather<<<dim3(100000), dim3(256), 0, stream>>>(dist_feat, ws + WS_BT, out_bond);
    rbf_wmma<13, 50><<<dim3(25000), dim3(64), 0, stream>>>(dist_feat,  bond_rbf_W,  bond_rbf_b,
                                                           out_bondf);
    rbf_wmma< 8, 32><<<dim3(25000), dim3(64), 0, stream>>>(angle_feat, angle_rbf_W, angle_rbf_b,
                                                           out_angle);
}